// _FilterAppend_40398462386567
// MI455X (gfx1250) — compile-verified
//
#include <hip/hip_runtime.h>

// Reference op:
//   x: (B=32, C=256, H=64, W=64) f32   y: (B=32, CATENDS=8) f32
//   out: (32, 264, 64, 64) f32 = concat([x, broadcast(y[:, :, None, None])], axis=1)
//
// Pure data movement: 134.2 MB read + 138.4 MB write ~= 272.6 MB HBM traffic,
// 0 FLOPs -> bandwidth bound, ~11.7 us floor @ 23.3 TB/s. No WMMA applicable.
//
// Round-1 change: amortize per-wave scheduling overhead.
//  - 1D grid: avoids the ~50-SALU ttmp6/ttmp7 cluster decode for blockIdx.y/z.
//  - 4 float4 per thread: one 256-thread block covers a whole 64x64 plane,
//    so each wave moves 2 KB (4x fewer waves, prologue amortized 4x).
//  - NT hints on all traffic: 272.6 MB streamed once, > 192 MB L2.

#define FA_B       32
#define FA_C       256
#define FA_CAT     8
#define FA_COUT    (FA_C + FA_CAT)   // 264
#define FA_HW4     1024              // float4 chunks per 64x64 plane
#define FA_VPT     4                 // float4s per thread (1024 / 256)

typedef float v4f __attribute__((ext_vector_type(4)));

__global__ __launch_bounds__(256)
void _FilterAppend_40398462386567_kernel(const v4f* __restrict__ x4,
                                         const float* __restrict__ y,
                                         v4f* __restrict__ out4)
{
    // grid.x = FA_B * FA_COUT = 8448 blocks; block = one (b, c) output plane.
    const int bc = blockIdx.x;            // b * 264 + c
    const int b  = bc / FA_COUT;          // scalar magic-multiply div
    const int c  = bc - b * FA_COUT;
    const int t  = threadIdx.x;           // 0..255

    const long long obase = ((long long)bc << 10);   // plane base in float4 units

    if (c < FA_C) {
        // Copy branch: out[b, c, :, :] = x[b, c, :, :]   (block-uniform branch)
        const long long ibase = ((long long)(b * FA_C + c)) << 10;
        #pragma unroll
        for (int k = 0; k < FA_VPT; ++k) {
            const int sp = t + k * 256;                          // 0..1023
            v4f v = __builtin_nontemporal_load(&x4[ibase + sp]); // global_load_b128 th:NT
            __builtin_nontemporal_store(v, &out4[obase + sp]);   // global_store_b128 th:NT
        }
    } else {
        // Broadcast branch: out[b, c, :, :] = y[b, c-256]
        // Address is block-uniform -> scalar load + lane broadcast.
        const float s = y[b * FA_CAT + (c - FA_C)];
        const v4f v = (v4f){s, s, s, s};
        #pragma unroll
        for (int k = 0; k < FA_VPT; ++k) {
            __builtin_nontemporal_store(v, &out4[obase + t + k * 256]);
        }
    }
}

extern "C" void kernel_launch(void* const* d_in, const int* in_sizes, int n_in,
                              void* d_out, int out_size, void* d_ws, size_t ws_size,
                              hipStream_t stream) {
    (void)in_sizes; (void)n_in; (void)out_size; (void)d_ws; (void)ws_size;

    const v4f*   x4 = (const v4f*)d_in[0];    // float32 x viewed as float4
    const float* y  = (const float*)d_in[1];  // float32 y
    v4f* out4       = (v4f*)d_out;            // float32 out viewed as float4

    dim3 grid(FA_B * FA_COUT);                // 8448 blocks, 1D
    dim3 block(256);                          // 8 wave32s
    _FilterAppend_40398462386567_kernel<<<grid, block, 0, stream>>>(x4, y, out4);
}